// BilinearUpsampling_88373247082947
// MI455X (gfx1250) — compile-verified
//
#include <hip/hip_runtime.h>

typedef __attribute__((ext_vector_type(2))) float v2f;
typedef __attribute__((ext_vector_type(8))) float v8f;

#define L2_EPS 1e-12f
#define C_DIM 128
#define CC_DIM (C_DIM * C_DIM)   // 16384

// One workgroup per (b,t) row. 256 threads = 8 wave32 waves.
// Wave i computes output tile-rows c in [16i, 16i+16), looping j over tile cols.
// Each 16x16 tile = (f*x_i) outer x_j via V_WMMA_F32_16X16X4_F32 (K=0 active).
__global__ __launch_bounds__(256) void outer_l2norm_upsample2_kernel(
    const float* __restrict__ x,   // (B*T, 128)
    const float* __restrict__ w,   // (1,)
    float* __restrict__ out,       // (B, 2T, 16384) flat
    int nRows)
{
    const int row = blockIdx.x;
    if (row >= nRows) return;

    __shared__ float xs[C_DIM];       // raw x row
    __shared__ float xsc[C_DIM];      // f * x row
    __shared__ float fshare;

    const int tid = threadIdx.x;
    const float* xrow = x + (size_t)row * C_DIM;

    if (tid < C_DIM) xs[tid] = xrow[tid];
    __syncthreads();

    // Wave 0: sum of squares over 128 elems, then scalar factor
    if (tid < 32) {
        float p = 0.f;
#pragma unroll
        for (int k = 0; k < 4; ++k) {
            float v = xs[tid + 32 * k];
            p += v * v;
        }
#pragma unroll
        for (int off = 16; off >= 1; off >>= 1)
            p += __shfl_xor(p, off, 32);
        if (tid == 0) {
            float wv = w[0];
            // sum(outer^2) = w^2 * s^2 ; normed = (w*x_c*x_d) * rsqrt(max(.,eps))
            float denom = sqrtf(fmaxf(wv * wv * p * p, L2_EPS));
            fshare = wv / denom;
        }
    }
    __syncthreads();

    const float f = fshare;
    if (tid < C_DIM) xsc[tid] = xs[tid] * f;
    __syncthreads();

    const int wave = tid >> 5;        // 0..7  -> tile row i
    const int lane = tid & 31;
    const int i = wave;

    // A fragment: 16x4 f32. Lanes 0-15: VGPR0=K0, VGPR1=K1; lanes 16-31: K2,K3.
    // Put f*x_i in K=0 column, zeros elsewhere -> D = (f*x_i) x_j^T exactly.
    v2f a;
    a.x = 0.f; a.y = 0.f;
    if (lane < 16) a.x = xsc[16 * i + lane];

    // Output bases for the two upsampled time slots
    float* o0 = out + (size_t)(2 * row) * CC_DIM;
    float* o1 = o0 + CC_DIM;

    const int mofs = (lane >> 4) << 3;   // C/D layout: lanes 16-31 hold M+8
    const int n = lane & 15;

    for (int j = 0; j < 8; ++j) {
        // B fragment: 4x16 f32, row K=0 striped over lanes 0-15 of VGPR0.
        v2f b;
        b.x = 0.f; b.y = 0.f;
        if (lane < 16) b.x = xs[16 * j + lane];

        v8f acc = {};
        acc = __builtin_amdgcn_wmma_f32_16x16x4_f32(
            /*neg_a=*/false, a, /*neg_b=*/false, b,
            /*c_mod=*/(short)0, acc, /*reuse_a=*/false, /*reuse_b=*/false);

        // D layout: VGPR r, lanes 0-15 -> (M=r, N=lane); lanes 16-31 -> (M=r+8, N=lane-16)
#pragma unroll
        for (int r = 0; r < 8; ++r) {
            const int cc = 16 * i + r + mofs;
            const int dd = 16 * j + n;
            const size_t off = (size_t)cc * C_DIM + dd;
            float v = acc[r];
            __builtin_nontemporal_store(v, o0 + off);   // t*2
            __builtin_nontemporal_store(v, o1 + off);   // t*2+1
        }
    }
}

extern "C" void kernel_launch(void* const* d_in, const int* in_sizes, int n_in,
                              void* d_out, int out_size, void* d_ws, size_t ws_size,
                              hipStream_t stream) {
    (void)n_in; (void)out_size; (void)d_ws; (void)ws_size;
    const float* x = (const float*)d_in[0];
    const float* w = (const float*)d_in[1];
    float* out = (float*)d_out;
    const int nRows = in_sizes[0] / C_DIM;   // B*T = 4096
    outer_l2norm_upsample2_kernel<<<nRows, 256, 0, stream>>>(x, w, out, nRows);
}